// DiffPool_4715874091424
// MI455X (gfx1250) — compile-verified
//
#include <hip/hip_runtime.h>
#include <hip/hip_bf16.h>
#include <math.h>

typedef float v2f __attribute__((ext_vector_type(2)));
typedef float v8f __attribute__((ext_vector_type(8)));

static constexpr int    NN    = 160;
static constexpr int    SLICE = NN * NN;            // 25600 elements per (b,c) slice
static constexpr int    BC    = 1024;               // B*C
static constexpr size_t HALF  = (size_t)BC * SLICE; // x_new | a_new split in d_out
static constexpr float  ALPHA_C = 0.05f;

// ---- WMMA f32 16x16x4 fragment helpers (ISA 7.12.2 layouts, wave32) ----
// A 16x4 (MxK): lanes 0-15 -> M=lane, VGPR0/1 = K,K+1 ; lanes 16-31 -> K+2,K+3
__device__ __forceinline__ v2f ldA(const float* A, int lda, int m0, int k, int lane) {
    int m  = m0 + (lane & 15);
    int kk = k + 2 * (lane >> 4);
    v2f f; f.x = A[m * lda + kk]; f.y = A[m * lda + kk + 1]; return f;
}
// A-operand = transpose of stored matrix S: A[m,k] = S[k,m]
__device__ __forceinline__ v2f ldAT(const float* S, int lds_, int m0, int k, int lane) {
    int m  = m0 + (lane & 15);
    int kk = k + 2 * (lane >> 4);
    v2f f; f.x = S[kk * lds_ + m]; f.y = S[(kk + 1) * lds_ + m]; return f;
}
// B 4x16 (KxN): lanes 0-15 -> N=lane, rows K / K+1 in VGPR0/1; lanes 16-31 -> K+2,K+3
__device__ __forceinline__ v2f ldB(const float* B, int ldb, int k, int n0, int lane) {
    int n  = n0 + (lane & 15);
    int kk = k + 2 * (lane >> 4);
    v2f f; f.x = B[kk * ldb + n]; f.y = B[(kk + 1) * ldb + n]; return f;
}
__device__ __forceinline__ v8f wmma4(v2f a, v2f b, v8f c) {
    return __builtin_amdgcn_wmma_f32_16x16x4_f32(false, a, false, b, (short)0, c, false, false);
}
// C/D 16x16: VGPR r -> row m0 + r + 8*lane[4], col n0 + lane[3:0]
__device__ __forceinline__ int crow(int m0, int r, int lane) { return m0 + r + 8 * (lane >> 4); }
__device__ __forceinline__ int ccol(int n0, int lane)        { return n0 + (lane & 15); }

// ---- CDNA5 async global->LDS staging (GLOBAL_LOAD_ASYNC_TO_LDS_B128, ASYNCcnt) ----
// Copies one SLICE tile (102400 B) into LDS at byte offset lds_off, 16 B per lane
// per issue, with no VGPR data staging. Caller drains with wait_async0() + barrier.
__device__ __forceinline__ void async_copy_tile(unsigned lds_off, const float* __restrict__ g,
                                                int tid, int nthreads) {
    for (int i = tid * 4; i < SLICE; i += nthreads * 4) {
        unsigned long long ga = (unsigned long long)(const void*)(g + i);
        unsigned lo = lds_off + (unsigned)i * 4u;
        asm volatile("global_load_async_to_lds_b128 %0, %1, off"
                     :: "v"(lo), "v"(ga) : "memory");
    }
}
__device__ __forceinline__ void wait_async0() {
    asm volatile("s_wait_asynccnt 0" ::: "memory");
}

// One wave computes a 16x160 output strip of a 160x160x160 GEMM, operands in LDS.
template <bool ATRANS>
__device__ __forceinline__ void gemm_strip(const float* Amat, const float* Bmat,
                                           int m0, int lane, v8f acc[10]) {
    const v8f z = {0.f, 0.f, 0.f, 0.f, 0.f, 0.f, 0.f, 0.f};
#pragma unroll
    for (int n = 0; n < 10; ++n) acc[n] = z;
    for (int k = 0; k < NN; k += 4) {
        v2f af = ATRANS ? ldAT(Amat, NN, m0, k, lane) : ldA(Amat, NN, m0, k, lane);
#pragma unroll
        for (int n = 0; n < 10; ++n) {
            v2f bf = ldB(Bmat, NN, k, 16 * n, lane);
            acc[n] = wmma4(af, bf, acc[n]);
        }
    }
}

// ---------------- K1: a1 = rownorm(a+I), a2 = rownorm(a^T+I) ----------------
__global__ void k_norm_adj(const float* __restrict__ a, float* __restrict__ a1,
                           float* __restrict__ a2) {
    int v = threadIdx.x;
    if (v >= NN) return;
    float rs = 1.0f, cs = 1.0f;
    for (int w = 0; w < NN; ++w) { rs += a[v * NN + w]; cs += a[w * NN + v]; }
    float ir = 1.0f / rs, ic = 1.0f / cs;
    for (int w = 0; w < NN; ++w) {
        float d = (v == w) ? 1.0f : 0.0f;
        a1[v * NN + w] = (a[v * NN + w] + d) * ir;
        a2[v * NN + w] = (a[w * NN + v] + d) * ic;
    }
}

// ---------------- K2: diffusion hops, accumulate hs1/hs2 over adjacencies ----
// Per block: one (b,c) slice. LDS holds A(100K) + X(100K) + H(100K) = 307KB.
__global__ void k_diffusion(const float* __restrict__ x, const float* __restrict__ aN,
                            float* __restrict__ hs1, float* __restrict__ hs2,
                            int accumulate) {
    extern __shared__ __align__(16) float lds[];
    float* A_s = lds;
    float* X_s = lds + SLICE;
    float* H_s = lds + 2 * SLICE;
    const int tid = threadIdx.x;
    const size_t off = (size_t)blockIdx.x * SLICE;

    const unsigned lbase = __builtin_amdgcn_groupstaticsize();
    async_copy_tile(lbase,                  aN,      tid, 320);
    async_copy_tile(lbase + SLICE * 4u,     x + off, tid, 320);
    wait_async0();
    __syncthreads();

    const int lane = tid & 31;
    const int m0   = (tid >> 5) * 16;   // 10 waves, one 16-row strip each
    v8f acc[10];

    // hop 1: H1 = alpha*X + (1-alpha) * A @ X
    gemm_strip<false>(A_s, X_s, m0, lane, acc);
#pragma unroll
    for (int n = 0; n < 10; ++n)
#pragma unroll
        for (int r = 0; r < 8; ++r) {
            int idx = crow(m0, r, lane) * NN + ccol(16 * n, lane);
            float h = ALPHA_C * X_s[idx] + (1.0f - ALPHA_C) * acc[n][r];
            H_s[idx] = h;
            if (accumulate) hs1[off + idx] += h; else hs1[off + idx] = h;
        }
    __syncthreads();

    // hop 2: H2 = alpha*X + (1-alpha) * A @ H1
    gemm_strip<false>(A_s, H_s, m0, lane, acc);
#pragma unroll
    for (int n = 0; n < 10; ++n)
#pragma unroll
        for (int r = 0; r < 8; ++r) {
            int idx = crow(m0, r, lane) * NN + ccol(16 * n, lane);
            float h = ALPHA_C * X_s[idx] + (1.0f - ALPHA_C) * acc[n][r];
            if (accumulate) hs2[off + idx] += h; else hs2[off + idx] = h;
        }
}

// ---------------- K3: fused channel mix: [We;Wp](64x96) @ [2x; hs1; hs2] -----
__device__ __forceinline__ float gval(const float* __restrict__ x,
                                      const float* __restrict__ h1,
                                      const float* __restrict__ h2,
                                      size_t bbase, int k, int nl) {
    if (k < 32) return 2.0f * x[bbase + (size_t)k * SLICE + nl];
    if (k < 64) return h1[bbase + (size_t)(k - 32) * SLICE + nl];
    return h2[bbase + (size_t)(k - 64) * SLICE + nl];
}

__global__ void k_mix(const float* __restrict__ x, const float* __restrict__ hs1,
                      const float* __restrict__ hs2, const float* __restrict__ We,
                      const float* __restrict__ be, const float* __restrict__ Wp,
                      const float* __restrict__ bp, float* __restrict__ out) {
    __shared__ float Wc[64 * 96];
    __shared__ float bc[64];
    const int tid = threadIdx.x;
    for (int i = tid; i < 64 * 96; i += blockDim.x) {
        int o = i / 96, k = i % 96;
        Wc[i] = (o < 32) ? We[o * 96 + k] : Wp[(o - 32) * 96 + k];
    }
    if (tid < 64) bc[tid] = 2.0f * ((tid < 32) ? be[tid] : bp[tid - 32]);
    __syncthreads();

    const int b    = blockIdx.y;
    const int lane = tid & 31;
    const int wv   = tid >> 5;                       // 8 waves
    const int nl0  = (blockIdx.x * 8 + wv) * 16;     // 16-wide column tile of N*L
    const size_t bbase = (size_t)b * 32 * SLICE;
    const int col = nl0 + (lane & 15);

    const v8f z = {0.f, 0.f, 0.f, 0.f, 0.f, 0.f, 0.f, 0.f};
    v8f acc[4] = {z, z, z, z};
    for (int k = 0; k < 96; k += 4) {
        int kk = k + 2 * (lane >> 4);
        v2f bf;
        bf.x = gval(x, hs1, hs2, bbase, kk,     col);
        bf.y = gval(x, hs1, hs2, bbase, kk + 1, col);
#pragma unroll
        for (int mt = 0; mt < 4; ++mt) {
            v2f af = ldA(Wc, 96, 16 * mt, k, lane);
            acc[mt] = wmma4(af, bf, acc[mt]);
        }
    }
#pragma unroll
    for (int mt = 0; mt < 4; ++mt)
#pragma unroll
        for (int r = 0; r < 8; ++r) {
            int o = 16 * mt + r + 8 * (lane >> 4);
            float v = acc[mt][r] + bc[o];
            if (o < 32)  out[(size_t)(b * 32 + o) * SLICE + col] = v;               // embed
            else         out[HALF + (size_t)(b * 32 + (o - 32)) * SLICE + col] = v; // pool
        }
}

// ---------------- K4: softmax over node axis (axis=2), slice-resident in LDS -
__global__ void k_softmax(float* __restrict__ sbuf) {
    extern __shared__ __align__(16) float P[];
    const size_t off = (size_t)blockIdx.x * SLICE;
    const int tid = threadIdx.x;   // 160 threads; thread = column l

    async_copy_tile(__builtin_amdgcn_groupstaticsize(), sbuf + off, tid, NN);
    wait_async0();
    __syncthreads();

    float m = -3.402823466e+38f;
    for (int n = 0; n < NN; ++n) m = fmaxf(m, P[n * NN + tid]);
    float s = 0.f;
    for (int n = 0; n < NN; ++n) s += __expf(P[n * NN + tid] - m);
    float inv = 1.0f / s;
    for (int n = 0; n < NN; ++n) P[n * NN + tid] = __expf(P[n * NN + tid] - m) * inv;
    __syncthreads();
    for (int i = tid * 4; i < SLICE; i += NN * 4)
        *(float4*)(sbuf + off + i) = *(const float4*)(P + i);
}

// ---------------- K5: x_new = s^T @ embed ; a_new = (s @ a) @ s --------------
// Loads embed/s slices fully into LDS first, so writing over them in d_out is safe.
__global__ void k_pool_apply(float* __restrict__ out, const float* __restrict__ araw) {
    extern __shared__ __align__(16) float lds[];
    float* S_s = lds;
    float* E_s = lds + SLICE;          // embed, later reused for t = s@a
    float* A_s = lds + 2 * SLICE;      // raw adjacency
    const int tid = threadIdx.x;
    const size_t off = (size_t)blockIdx.x * SLICE;

    const unsigned lbase = __builtin_amdgcn_groupstaticsize();
    async_copy_tile(lbase,                  out + HALF + off, tid, 320); // S
    async_copy_tile(lbase + SLICE * 4u,     out + off,        tid, 320); // E
    async_copy_tile(lbase + 2u * SLICE * 4u, araw,            tid, 320); // A
    wait_async0();
    __syncthreads();

    const int lane = tid & 31;
    const int m0   = (tid >> 5) * 16;
    v8f acc[10];

    // x_new[l1,l2] = sum_n S[n,l1] * E[n,l2]  (A-operand = S^T)
    gemm_strip<true>(S_s, E_s, m0, lane, acc);
#pragma unroll
    for (int n = 0; n < 10; ++n)
#pragma unroll
        for (int r = 0; r < 8; ++r)
            out[off + crow(m0, r, lane) * NN + ccol(16 * n, lane)] = acc[n][r];
    __syncthreads();

    // t = S @ A  -> reuse E_s
    gemm_strip<false>(S_s, A_s, m0, lane, acc);
#pragma unroll
    for (int n = 0; n < 10; ++n)
#pragma unroll
        for (int r = 0; r < 8; ++r)
            E_s[crow(m0, r, lane) * NN + ccol(16 * n, lane)] = acc[n][r];
    __syncthreads();

    // a_new = t @ S
    gemm_strip<false>(E_s, S_s, m0, lane, acc);
#pragma unroll
    for (int n = 0; n < 10; ++n)
#pragma unroll
        for (int r = 0; r < 8; ++r)
            out[HALF + off + crow(m0, r, lane) * NN + ccol(16 * n, lane)] = acc[n][r];
}

extern "C" void kernel_launch(void* const* d_in, const int* in_sizes, int n_in,
                              void* d_out, int out_size, void* d_ws, size_t ws_size,
                              hipStream_t stream) {
    const float* x  = (const float*)d_in[0];
    const float* a  = (const float*)d_in[1];
    const float* We = (const float*)d_in[2];
    const float* be = (const float*)d_in[3];
    const float* Wp = (const float*)d_in[4];
    const float* bp = (const float*)d_in[5];
    float* out = (float*)d_out;
    float* ws  = (float*)d_ws;

    float* a1  = ws;                         // 25600
    float* a2  = ws + SLICE;                 // 25600
    float* hs1 = ws + 2 * SLICE;             // 26.2M
    float* hs2 = hs1 + (size_t)BC * SLICE;   // 26.2M   (~210 MB total scratch)

    const size_t lds3 = (size_t)3 * SLICE * sizeof(float);   // 307200 B (<320KB WGP LDS)

    k_norm_adj<<<1, NN, 0, stream>>>(a, a1, a2);
    k_diffusion<<<BC, 320, lds3, stream>>>(x, a1, hs1, hs2, 0);
    k_diffusion<<<BC, 320, lds3, stream>>>(x, a2, hs1, hs2, 1);
    k_mix<<<dim3(200, 32), 256, 0, stream>>>(x, hs1, hs2, We, be, Wp, bp, out);
    k_softmax<<<BC, NN, (size_t)SLICE * sizeof(float), stream>>>(out + HALF);
    k_pool_apply<<<BC, 320, lds3, stream>>>(out, a);
}